// GraphNN_80075370266804
// MI455X (gfx1250) — compile-verified
//
#include <hip/hip_runtime.h>

#define N_NODES 100000
#define N_EDGES 1200000
#define D_IN 64
#define D_HID 128

typedef __attribute__((ext_vector_type(2))) float v2f;
typedef __attribute__((ext_vector_type(8))) float v8f;

// ---------------------------------------------------------------------------
// Zero-fill (float4 stores)
// ---------------------------------------------------------------------------
__global__ void zero_kernel(float4* __restrict__ p, int n4) {
    int i = blockIdx.x * blockDim.x + threadIdx.x;
    if (i < n4) p[i] = make_float4(0.f, 0.f, 0.f, 0.f);
}

// ---------------------------------------------------------------------------
// Edge scatter-add: agg[dst[e], :] += feat[src[e], :]
// One wave32 per edge; lanes cover the feature dim (coalesced 128B groups).
// f32 hardware atomics resolve in L2 (dest accumulators fit in 192MB L2).
// ---------------------------------------------------------------------------
template <int D>
__global__ void scatter_add_kernel(const float* __restrict__ feat,
                                   const int* __restrict__ src,
                                   const int* __restrict__ dst,
                                   float* __restrict__ agg,
                                   int n_edges) {
    const int lane = threadIdx.x & 31;
    const int wave = blockIdx.x * (blockDim.x >> 5) + (threadIdx.x >> 5);
    const int stride = gridDim.x * (blockDim.x >> 5);
    for (int e = wave; e < n_edges; e += stride) {
        const int s = src[e];
        const int d = dst[e];
        const float* fs = feat + (size_t)s * D;
        float* fd = agg + (size_t)d * D;
#pragma unroll
        for (int t = 0; t < D / 32; ++t) {
            float v = fs[lane + 32 * t];
            unsafeAtomicAdd(&fd[lane + 32 * t], v);   // -> global_atomic_add_f32
        }
    }
}

// ---------------------------------------------------------------------------
// Fused dual GEMM via V_WMMA_F32_16X16X4_F32:
//   D[m, n] = sum_k A1[m,k] * W1[n,k]  +  sum_k A2[m,k] * W2[n,k]  + bias[n]
// (optionally ReLU'd). W matrices are [D_HID x K] row-major, used transposed.
// One wave per 16x16 output tile; block = 8 waves = one M-tile x all 8 N-tiles
// so the 8 waves share A rows through L0/L2.
//
// VGPR layouts per CDNA5 ISA 7.12.2 (f32, 16x16x4):
//   A: lane<16 -> A[m][k+0..1], lane>=16 -> A[m][k+2..3]   (m = lane&15)
//   B: lane<16 -> B[k+0..1][n], lane>=16 -> B[k+2..3][n]   (n = lane&15)
//   D: vgpr r  -> row 8*(lane>>4)+r, col lane&15
// ---------------------------------------------------------------------------
__global__ __launch_bounds__(256) void gemm_dual_wmma(
        const float* __restrict__ A1, int K1, const float* __restrict__ W1,
        const float* __restrict__ A2, int K2, const float* __restrict__ W2,
        const float* __restrict__ bias, float* __restrict__ D,
        int M, int do_relu) {
    const int tile_m = blockIdx.x;          // 16 output rows per block
    const int tile_n = threadIdx.x >> 5;    // wave id in block -> 16-col tile
    if (tile_m * 16 >= M) return;           // wave-uniform guard (EXEC stays full)

    const int lane = threadIdx.x & 31;
    const int half = lane >> 4;             // K sub-offset selector
    const int mn   = lane & 15;

    const int row = tile_m * 16 + mn;       // A row handled by this lane
    const int col = tile_n * 16 + mn;       // output column / weight row

    v8f c = {};

    const float* a_ptr = A1 + (size_t)row * K1 + 2 * half;
    const float* w_ptr = W1 + (size_t)col * K1 + 2 * half;
    for (int k = 0; k < K1; k += 4) {
        v2f a = *(const v2f*)(a_ptr + k);
        v2f b = *(const v2f*)(w_ptr + k);
        c = __builtin_amdgcn_wmma_f32_16x16x4_f32(false, a, false, b,
                                                  (short)0, c, false, false);
    }

    a_ptr = A2 + (size_t)row * K2 + 2 * half;
    w_ptr = W2 + (size_t)col * K2 + 2 * half;
    for (int k = 0; k < K2; k += 4) {
        v2f a = *(const v2f*)(a_ptr + k);
        v2f b = *(const v2f*)(w_ptr + k);
        c = __builtin_amdgcn_wmma_f32_16x16x4_f32(false, a, false, b,
                                                  (short)0, c, false, false);
    }

    const int out_col = tile_n * 16 + mn;
    const float bv = bias[out_col];
#pragma unroll
    for (int r = 0; r < 8; ++r) {
        const int out_row = tile_m * 16 + 8 * half + r;
        float v = c[r] + bv;
        if (do_relu) v = fmaxf(v, 0.f);
        D[(size_t)out_row * D_HID + out_col] = v;
    }
}

// ---------------------------------------------------------------------------
// Launch: agg1 -> layer1 (relu) -> agg2 -> layer2
// ws layout: [ h : N*128 f32 ][ agg : N*128 f32 (layer1 uses first N*64) ]
// ---------------------------------------------------------------------------
extern "C" void kernel_launch(void* const* d_in, const int* in_sizes, int n_in,
                              void* d_out, int out_size, void* d_ws, size_t ws_size,
                              hipStream_t stream) {
    const float* x       = (const float*)d_in[0];
    const int*   edge    = (const int*)d_in[1];   // [2, E] int32
    const float* W1_rel  = (const float*)d_in[2];
    const float* b1_rel  = (const float*)d_in[3];
    const float* W1_root = (const float*)d_in[4];
    const float* W2_rel  = (const float*)d_in[5];
    const float* b2_rel  = (const float*)d_in[6];
    const float* W2_root = (const float*)d_in[7];
    float* out = (float*)d_out;

    const int* src = edge;            // row 0
    const int* dst = edge + N_EDGES;  // row 1

    float* h   = (float*)d_ws;                       // N*128 f32
    float* agg = h + (size_t)N_NODES * D_HID;        // N*128 f32 scratch

    const int scatter_blocks = N_EDGES / 8;          // 8 waves/block, 1 edge/wave
    const int gemm_blocks    = N_NODES / 16;         // 16 rows/block (exact)

    // ---- layer 1 ----
    {
        int n4 = N_NODES * D_IN / 4;
        zero_kernel<<<(n4 + 255) / 256, 256, 0, stream>>>((float4*)agg, n4);
    }
    scatter_add_kernel<D_IN><<<scatter_blocks, 256, 0, stream>>>(x, src, dst, agg, N_EDGES);
    gemm_dual_wmma<<<gemm_blocks, 256, 0, stream>>>(agg, D_IN, W1_rel,
                                                    x, D_IN, W1_root,
                                                    b1_rel, h, N_NODES, 1);

    // ---- layer 2 ----
    {
        int n4 = N_NODES * D_HID / 4;
        zero_kernel<<<(n4 + 255) / 256, 256, 0, stream>>>((float4*)agg, n4);
    }
    scatter_add_kernel<D_HID><<<scatter_blocks, 256, 0, stream>>>(h, src, dst, agg, N_EDGES);
    gemm_dual_wmma<<<gemm_blocks, 256, 0, stream>>>(agg, D_HID, W2_rel,
                                                    h, D_HID, W2_root,
                                                    b2_rel, out, N_NODES, 0);
}